// FastHebbMemory_57543971832106
// MI455X (gfx1250) — compile-verified
//
#include <hip/hip_runtime.h>
#include <math.h>
#include <stdint.h>

#define N_ROWS 131072
#define DIM    1024
#define TOPK   32
#define TAU    0.2f
#define EPSN   1e-12f

#define WAVES    8        // waves per GEMV block (256 threads)
#define CHUNK2   32       // k-chunk staged in LDS per wave (double-buffered)
#define NCHUNK   (DIM / CHUNK2)
#define LSTRIDE2 36       // 32 + 4 pad: 16B-aligned rows, conflict-free column reads

#define P1_BLOCKS 128     // top-k phase 1 blocks
#define P1_CHUNK  1024    // elements per phase-1 block
#define NCAND     (P1_BLOCKS * TOPK)   // 4096 candidates

typedef float v2f __attribute__((ext_vector_type(2)));
typedef float v8f __attribute__((ext_vector_type(8)));

// ---------------------------------------------------------------------------
// Kernel 1: qs = query / (max(||query||, eps) * tau)   (folds 1/tau into q)
// ---------------------------------------------------------------------------
__global__ __launch_bounds__(256) void fh_normalize(const float* __restrict__ q,
                                                    float* __restrict__ qs) {
  __shared__ float red[256];
  __shared__ float scale;
  const int tid = threadIdx.x;
  float ss = 0.f;
  for (int d = tid; d < DIM; d += 256) { const float v = q[d]; ss += v * v; }
  red[tid] = ss;
  __syncthreads();
  for (int s = 128; s > 0; s >>= 1) {
    if (tid < s) red[tid] += red[tid + s];
    __syncthreads();
  }
  if (tid == 0) {
    float n = sqrtf(red[0]);
    n = fmaxf(n, EPSN);
    scale = 1.0f / (n * TAU);
  }
  __syncthreads();
  for (int d = tid; d < DIM; d += 256) qs[d] = q[d] * scale;
}

// ---------------------------------------------------------------------------
// Async-stage one 16 x CHUNK2 K-tile into an LDS buffer (4 b128 per lane).
// LDS dest address per lane goes in the async VDST operand; low 32 bits of a
// flat pointer to __shared__ are the LDS byte offset (aperture rule).
// ---------------------------------------------------------------------------
__device__ __forceinline__ void fh_issue_async_chunk(const float* __restrict__ K,
                                                     int rowb, int kb,
                                                     float* tilebuf,
                                                     int r8, int c8) {
#pragma unroll
  for (int it = 0; it < 4; ++it) {
    const int rr = it * 4 + r8;
    const uint64_t ga = (uint64_t)(uintptr_t)(K + (size_t)(rowb + rr) * DIM + kb + c8);
    const uint32_t la = (uint32_t)(uintptr_t)(tilebuf + rr * LSTRIDE2 + c8);
    asm volatile("global_load_async_to_lds_b128 %0, %1, off"
                 :: "v"(la), "v"(ga)
                 : "memory");
  }
}

// ---------------------------------------------------------------------------
// Kernel 2: logits = K @ qs via V_WMMA_F32_16X16X4_F32.
// Each wave computes 16 rows; K staged via async global->LDS, double-buffered.
// ---------------------------------------------------------------------------
__global__ __launch_bounds__(256) void fh_gemv_wmma(const float* __restrict__ K,
                                                    const float* __restrict__ qs,
                                                    float* __restrict__ logits) {
  __shared__ float qsl[DIM];                            // 4 KB
  __shared__ float tile[WAVES][2][16 * LSTRIDE2];       // 36 KB

  const int tid  = threadIdx.x;
  const int w    = tid >> 5;       // wave id (wave32)
  const int lane = tid & 31;
  const int half = lane >> 4;      // selects K-slot pair {0,1} vs {2,3}
  const int m    = lane & 15;      // row within 16-row tile
  const int r8   = lane >> 3;      // staging: row group 0..3
  const int c8   = (lane & 7) * 4; // staging: float4 column 0..28

  for (int d = tid; d < DIM; d += 256) qsl[d] = qs[d];
  __syncthreads();

  const int rowb = blockIdx.x * (WAVES * 16) + w * 16;

  v8f acc = {0.f, 0.f, 0.f, 0.f, 0.f, 0.f, 0.f, 0.f};

  int buf = 0;
  fh_issue_async_chunk(K, rowb, 0, &tile[w][0][0], r8, c8);

  for (int i = 0; i < NCHUNK; ++i) {
    const int kb = i * CHUNK2;
    if (i + 1 < NCHUNK) {
      fh_issue_async_chunk(K, rowb, kb + CHUNK2, &tile[w][buf ^ 1][0], r8, c8);
      // in-order completion: <=4 outstanding -> current buffer has landed
      asm volatile("s_wait_asynccnt 0x4" ::: "memory");
    } else {
      asm volatile("s_wait_asynccnt 0x0" ::: "memory");
    }

    // ---- 8 WMMAs per chunk: A[m,k]=K[rowb+m][kb+kk+k], B[k,n]=qs[kb+kk+k] ----
    const float* arow = &tile[w][buf][m * LSTRIDE2 + 2 * half];
    const float* brow = &qsl[kb + 2 * half];
#pragma unroll
    for (int kk = 0; kk < CHUNK2; kk += 4) {
      v2f a = *(const v2f*)(arow + kk);   // A vgpr j <- K[.., kk + j + 2*half]
      v2f b = *(const v2f*)(brow + kk);   // B vgpr j <- qs[kk + j + 2*half] (bcast cols)
      acc = __builtin_amdgcn_wmma_f32_16x16x4_f32(
          false, a, false, b, (short)0, acc, false, false);
    }
    buf ^= 1;
  }

  // C/D layout: acc[j] = C[j + 8*half, n] (same value in all 16 lanes of half)
  if ((lane & 15) == 0) {
#pragma unroll
    for (int jv = 0; jv < 8; ++jv)
      logits[rowb + 8 * half + jv] = acc[jv];
  }
}

// ---------------------------------------------------------------------------
// Kernel 3: zero full_attn
// ---------------------------------------------------------------------------
__global__ void fh_zero(float* __restrict__ p, int n) {
  const int i = blockIdx.x * blockDim.x + threadIdx.x;
  if (i < n) p[i] = 0.f;
}

// ---------------------------------------------------------------------------
// Kernel 4 (phase 1): per-chunk local top-32 -> candidate (val, idx) lists
// ---------------------------------------------------------------------------
__global__ __launch_bounds__(256) void fh_topk_p1(const float* __restrict__ logits,
                                                  float* __restrict__ cand_val,
                                                  int* __restrict__ cand_idx) {
  __shared__ float lv[P1_CHUNK];
  __shared__ float rv[256];
  __shared__ int   ri[256];
  const int tid  = threadIdx.x;
  const int base = blockIdx.x * P1_CHUNK;

  for (int i = tid; i < P1_CHUNK; i += 256) lv[i] = logits[base + i];
  __syncthreads();

  for (int t = 0; t < TOPK; ++t) {
    float bv = -INFINITY; int bi = -1;
    for (int i = tid; i < P1_CHUNK; i += 256) {
      const float v = lv[i];
      if (v > bv || (v == bv && i < bi)) { bv = v; bi = i; }
    }
    rv[tid] = bv; ri[tid] = bi;
    __syncthreads();
    for (int s = 128; s > 0; s >>= 1) {
      if (tid < s) {
        const float ov = rv[tid + s]; const int oi = ri[tid + s];
        if (ov > rv[tid] ||
            (ov == rv[tid] && oi >= 0 && (ri[tid] < 0 || oi < ri[tid]))) {
          rv[tid] = ov; ri[tid] = oi;
        }
      }
      __syncthreads();
    }
    if (tid == 0) {
      cand_val[blockIdx.x * TOPK + t] = rv[0];
      cand_idx[blockIdx.x * TOPK + t] = base + ri[0];
      lv[ri[0]] = -INFINITY;
    }
    __syncthreads();
  }
}

// ---------------------------------------------------------------------------
// Kernel 5 (phase 2): global top-32 of candidates, softmax, scatter, V-sum
// ---------------------------------------------------------------------------
__global__ __launch_bounds__(1024) void fh_topk_p2(const float* __restrict__ cand_val,
                                                   const int* __restrict__ cand_idx,
                                                   const float* __restrict__ V,
                                                   float* __restrict__ out,
                                                   float* __restrict__ full_attn) {
  __shared__ float cv[NCAND];
  __shared__ int   ci[NCAND];
  __shared__ float rv[1024];
  __shared__ int   ri[1024];
  __shared__ float svals[TOPK];
  __shared__ int   sidx[TOPK];
  __shared__ float sattn[TOPK];
  __shared__ float sinv;

  const int tid = threadIdx.x;
  for (int i = tid; i < NCAND; i += 1024) { cv[i] = cand_val[i]; ci[i] = cand_idx[i]; }
  __syncthreads();

  for (int t = 0; t < TOPK; ++t) {
    float bv = -INFINITY; int bi = -1, bgi = 0x7fffffff;
    for (int i = tid; i < NCAND; i += 1024) {
      const float v = cv[i];
      if (v > bv || (v == bv && ci[i] < bgi)) { bv = v; bi = i; bgi = ci[i]; }
    }
    rv[tid] = bv; ri[tid] = bi;
    __syncthreads();
    for (int s = 512; s > 0; s >>= 1) {
      if (tid < s) {
        const float ov = rv[tid + s]; const int oi = ri[tid + s];
        if (oi >= 0 &&
            (ri[tid] < 0 || ov > rv[tid] ||
             (ov == rv[tid] && ci[oi] < ci[ri[tid]]))) {
          rv[tid] = ov; ri[tid] = oi;
        }
      }
      __syncthreads();
    }
    if (tid == 0) {
      svals[t] = rv[0];
      sidx[t]  = ci[ri[0]];
      cv[ri[0]] = -INFINITY;
    }
    __syncthreads();
  }

  // softmax over the 32 (descending, max is svals[0])
  if (tid < TOPK) sattn[tid] = expf(svals[tid] - svals[0]);
  __syncthreads();
  if (tid == 0) {
    float sum = 0.f;
    for (int i = 0; i < TOPK; ++i) sum += sattn[i];
    sinv = 1.0f / sum;
  }
  __syncthreads();
  if (tid < TOPK) {
    const float a = sattn[tid] * sinv;
    sattn[tid] = a;
    full_attn[sidx[tid]] = a;       // scatter (full_attn pre-zeroed)
  }
  __syncthreads();

  // out[d] = sum_i attn[i] * V[idx[i], d]   (coalesced over d = tid)
  float accum = 0.f;
#pragma unroll 4
  for (int i = 0; i < TOPK; ++i)
    accum += sattn[i] * V[(size_t)sidx[i] * DIM + tid];
  out[tid] = accum;
}

// ---------------------------------------------------------------------------
extern "C" void kernel_launch(void* const* d_in, const int* in_sizes, int n_in,
                              void* d_out, int out_size, void* d_ws, size_t ws_size,
                              hipStream_t stream) {
  (void)in_sizes; (void)n_in; (void)out_size; (void)ws_size;

  const float* query = (const float*)d_in[0];
  const float* K     = (const float*)d_in[1];
  const float* V     = (const float*)d_in[2];
  // d_in[3] = topk (== 32, compile-time TOPK)

  float* out       = (float*)d_out;        // [DIM]
  float* full_attn = out + DIM;            // [N_ROWS]

  float* qs       = (float*)d_ws;          // [DIM]
  float* logits   = qs + DIM;              // [N_ROWS]
  float* cand_val = logits + N_ROWS;       // [NCAND]
  int*   cand_idx = (int*)(cand_val + NCAND); // [NCAND]

  fh_normalize<<<1, 256, 0, stream>>>(query, qs);
  fh_gemv_wmma<<<N_ROWS / (WAVES * 16), 256, 0, stream>>>(K, qs, logits);
  fh_zero<<<N_ROWS / 256, 256, 0, stream>>>(full_attn, N_ROWS);
  fh_topk_p1<<<P1_BLOCKS, 256, 0, stream>>>(logits, cand_val, cand_idx);
  fh_topk_p2<<<1, 1024, 0, stream>>>(cand_val, cand_idx, V, out, full_attn);
}